// OrderedRoISampling_49108656062899
// MI455X (gfx1250) — compile-verified
//
#include <hip/hip_runtime.h>
#include <hip/hip_bf16.h>

#define HF  32   // feature map H
#define WFD 32   // feature map W
#define CCH 64   // channels
#define HO  8    // output bins per axis

typedef __attribute__((ext_vector_type(2))) float v2f;
typedef __attribute__((ext_vector_type(8))) float v8f;

// ---------------------------------------------------------------------------
// Kernel 1: per-box separable bilinear weight matrices.
//   Wy[n][oy][y] (scaled by 1/count) and Wx[n][ox][x], plus flip code.
//   One 64-thread block per box; one (axis, bin, sample) per thread;
//   scatter-add of the two bilinear taps via LDS float atomics (ds_add_f32).
// ---------------------------------------------------------------------------
__global__ __launch_bounds__(64) void roi_weights_kernel(
    const int* __restrict__ starts, const int* __restrict__ goals,
    float* __restrict__ wy, float* __restrict__ wx, int* __restrict__ flip)
{
  __shared__ float Wl[2][HO][WFD];   // [axis][bin][src coord]
  const int n = blockIdx.x;
  const int t = threadIdx.x;

  for (int r = t; r < 2 * HO * WFD; r += 64) ((float*)Wl)[r] = 0.0f;
  __syncthreads();

  const float s0 = (float)starts[2*n+0], s1 = (float)starts[2*n+1];
  const float g0 = (float)goals [2*n+0], g1 = (float)goals [2*n+1];
  const float mn0 = fminf(s0, g0), mx0 = fmaxf(s0, g0);
  const float mn1 = fminf(s1, g1), mx1 = fmaxf(s1, g1);

  const int axis = t >> 5;        // 0 = y, 1 = x
  const int idx  = t & 31;
  const int oy   = idx >> 2;      // output bin 0..7
  const int i    = idx & 3;       // adaptive sample 0..3 (MAXG)

  const float mn  = axis ? mn1 : mn0;
  const float mx  = axis ? mx1 : mx0;
  const float bin = (mx - mn) * 0.125f;         // roi/8 (exact pow2 scale)
  const int   gr  = (int)ceilf(bin);            // adaptive grid, <= 4
  const float gs  = (float)((gr > 1) ? gr : 1);
  const float pos = (mn - 0.5f) + (float)oy * bin + ((float)i + 0.5f) * (bin / gs);
  const bool  valid = (pos >= -1.0f) && (pos <= 32.0f) && (i < gr);
  const float p   = fminf(fmaxf(pos, 0.0f), 31.0f);
  const int   lo  = (int)floorf(p);
  const int   hi  = (lo >= 31) ? 31 : lo + 1;
  const float l   = p - (float)lo;
  const float vf  = valid ? 1.0f : 0.0f;

  atomicAdd(&Wl[axis][oy][lo], (1.0f - l) * vf);
  atomicAdd(&Wl[axis][oy][hi], l * vf);
  __syncthreads();

  const int gh = (int)ceilf((mx0 - mn0) * 0.125f);
  const int gw = (int)ceilf((mx1 - mn1) * 0.125f);
  int cnt = gh * gw; if (cnt < 1) cnt = 1;
  const float inv = 1.0f / (float)cnt;          // folded into Wy

  for (int r = t; r < 2 * HO * WFD; r += 64) {
    const float v = ((float*)Wl)[r];
    if (r < HO * WFD) wy[n * (HO*WFD) + r] = v * inv;
    else              wx[n * (HO*WFD) + (r - HO*WFD)] = v;
  }
  if (t == 0)
    flip[n] = ((s0 > g0) ? 2 : 0) + ((s1 > g1) ? 1 : 0);
}

// ---------------------------------------------------------------------------
// Kernel 2: per-box R = (Wy/count) @ F[b,c] @ Wx^T via V_WMMA_F32_16X16X4_F32.
//   128 threads = 4 wave32 per box; each wave owns 16 channels.
//   GEMM1 (8x32x32, M zero-padded to 16): 2 N-tiles x 8 K-steps of WMMA.
//   T spilled to per-wave LDS tile; GEMM2 (8x8x32) + flip fold done in VALU.
// ---------------------------------------------------------------------------
__global__ __launch_bounds__(128) void roi_gemm_kernel(
    const float* __restrict__ feats, const int* __restrict__ batch_idxs,
    const float* __restrict__ wy, const float* __restrict__ wx,
    const int* __restrict__ flip, float* __restrict__ out)
{
  __shared__ float sWy[HO][WFD];
  __shared__ float sWx[HO][WFD];
  __shared__ float sT[4][WFD][HO];   // [wave][x][oy]

  const int n    = blockIdx.x;
  const int tid  = threadIdx.x;
  const int wave = tid >> 5;
  const int lane = tid & 31;

  for (int r = tid; r < HO * WFD; r += 128) {
    ((float*)sWy)[r] = wy[n * (HO*WFD) + r];
    ((float*)sWx)[r] = wx[n * (HO*WFD) + r];
  }
  __syncthreads();

  const int b  = batch_idxs[n];
  const int fc = flip[n];

  // --- A-operand (16x4 f32 per ISA layout): lanes 0-15 hold K={0,1},
  //     lanes 16-31 hold K={2,3}; rows m>=8 are zero padding.
  const int m    = lane & 15;
  const int koff = (lane < 16) ? 0 : 2;
  float a0[8], a1[8];
#pragma unroll
  for (int s = 0; s < 8; ++s) {
    a0[s] = (m < 8) ? sWy[m][4*s + koff]     : 0.0f;
    a1[s] = (m < 8) ? sWy[m][4*s + koff + 1] : 0.0f;
  }

  // --- per-lane output->R mapping with flip folded in (reference table bug kept):
  //     fc 0: R[j][k], fc 3: R[7-j][k], fc 1/2: R[j][7-j]
  int pa[2], pb[2];
#pragma unroll
  for (int h = 0; h < 2; ++h) {
    const int p = lane + 32 * h;
    const int j = p >> 3, k = p & 7;
    int aa, bb;
    if (fc == 0)      { aa = j;     bb = k;     }
    else if (fc == 3) { aa = 7 - j; bb = k;     }
    else              { aa = j;     bb = 7 - j; }
    pa[h] = aa; pb[h] = bb;
  }

  const int xcol = lane & 15;
  for (int cc = 0; cc < 16; ++cc) {
    const int c = wave * 16 + cc;
    const float* F = feats + (((size_t)b * CCH + c) * (HF * WFD));

    v8f acc0 = {0.f,0.f,0.f,0.f,0.f,0.f,0.f,0.f};
    v8f acc1 = {0.f,0.f,0.f,0.f,0.f,0.f,0.f,0.f};
#pragma unroll
    for (int s = 0; s < 8; ++s) {
      const int r0 = 4*s + koff;            // B 4x16 row-striped: K={0,2}/{1,3}
      v2f A  = { a0[s], a1[s] };
      v2f B0 = { F[r0*WFD + xcol],      F[(r0+1)*WFD + xcol]      };
      v2f B1 = { F[r0*WFD + 16 + xcol], F[(r0+1)*WFD + 16 + xcol] };
      acc0 = __builtin_amdgcn_wmma_f32_16x16x4_f32(false, A, false, B0,
                                                   (short)0, acc0, false, false);
      acc1 = __builtin_amdgcn_wmma_f32_16x16x4_f32(false, A, false, B1,
                                                   (short)0, acc1, false, false);
    }

    // D layout: vgpr v, lanes 0-15 => row M=v (our valid rows 0..7), col=lane.
    if (lane < 16) {
#pragma unroll
      for (int v = 0; v < 8; ++v) {
        sT[wave][lane][v]      = acc0[v];   // x = lane
        sT[wave][lane + 16][v] = acc1[v];   // x = lane + 16
      }
    }
    __syncthreads();

    // GEMM2 + flip: each lane produces 2 of the 64 outputs of this channel.
    float r[2];
#pragma unroll
    for (int h = 0; h < 2; ++h) {
      float ssum = 0.0f;
#pragma unroll
      for (int x = 0; x < WFD; ++x)
        ssum = fmaf(sT[wave][x][pa[h]], sWx[pb[h]][x], ssum);
      r[h] = ssum;
    }
    float* O = out + (((size_t)n * CCH + c) << 6);
    O[lane]      = r[0];   // coalesced 128B
    O[lane + 32] = r[1];
    __syncthreads();
  }
}

// ---------------------------------------------------------------------------
extern "C" void kernel_launch(void* const* d_in, const int* in_sizes, int n_in,
                              void* d_out, int out_size, void* d_ws, size_t ws_size,
                              hipStream_t stream) {
  const float* feats      = (const float*)d_in[0];
  const int*   batch_idxs = (const int*)  d_in[1];
  const int*   starts     = (const int*)  d_in[2];
  const int*   goals      = (const int*)  d_in[3];
  const int N = in_sizes[1];

  float* wy   = (float*)d_ws;                      // N * 256 floats
  float* wx   = wy + (size_t)N * (HO * WFD);       // N * 256 floats
  int*   flip = (int*)(wx + (size_t)N * (HO * WFD));

  roi_weights_kernel<<<N, 64, 0, stream>>>(starts, goals, wy, wx, flip);
  roi_gemm_kernel<<<N, 128, 0, stream>>>(feats, batch_idxs, wy, wx, flip,
                                         (float*)d_out);
}